// DilatedKnnGraph_5549097746952
// MI455X (gfx1250) — compile-verified
//
#include <hip/hip_runtime.h>

// DilatedKnnGraph on gfx1250 (MI455X, wave32).
// d(m,n) = |x_m|^2 + |x_n|^2 - 2 <x_m, x_n>.  For per-row top-k the |x_m|^2
// term is a row-constant and is dropped; key(m,n) = |x_n|^2 - 2 G(m,n) with
// G computed 16x16-tile-at-a-time via V_WMMA_F32_16X16X4_F32 (C=3 padded to K=4).
// v2: software-pipelined candidate loads + tile-level min guard for the top-k filter.

typedef __attribute__((ext_vector_type(2))) float v2f;
typedef __attribute__((ext_vector_type(8))) float v8f;

#define NPTS 8192
#define KSEL 16
#define KOUT 8
#define ROWS_PER_WAVE 32          // two 16-row WMMA M-tiles per wave
#define WAVES_PER_BLOCK 8
#define ROWS_PER_BLOCK (ROWS_PER_WAVE * WAVES_PER_BLOCK)   // 256

__global__ __launch_bounds__(256) void dilated_knn_wmma_kernel(
    const float* __restrict__ x,   // (4, 8192, 3) f32
    int* __restrict__ out)         // (4, 8192, 8) i32
{
    // Per-wave transpose buffer: 32 rows x 16 cols of f32 keys (2 KB/wave).
    __shared__ float keybuf[WAVES_PER_BLOCK][ROWS_PER_WAVE * 16];

    const int tid  = threadIdx.x;
    const int w    = tid >> 5;
    const int lane = tid & 31;
    const int nc   = lane & 15;    // column (N) / row (M) sub-index
    const int hi   = lane >> 4;    // 0: K=0,1 half ; 1: K=2,3 half

    const int growbase = blockIdx.x * ROWS_PER_BLOCK + w * ROWS_PER_WAVE;
    const int b  = growbase / NPTS;        // batch (blocks never straddle batches)
    const int r0 = growbase % NPTS;        // first row of this wave within batch
    const float* __restrict__ xb = x + (size_t)b * NPTS * 3;

    // ---- Loop-invariant A fragments (16x4 f32 layout):
    // lanes 0-15:  VGPR0 = A[M=nc][K=0], VGPR1 = A[M=nc][K=1]
    // lanes 16-31: VGPR0 = A[M=nc][K=2], VGPR1 = A[M=nc][K=3] (= 0 pad)
    v2f afrag[2];
#pragma unroll
    for (int t = 0; t < 2; ++t) {
        const int row = r0 + 16 * t + nc;
        const float a0 = xb[row * 3 + 0];
        const float a1 = xb[row * 3 + 1];
        const float a2 = xb[row * 3 + 2];
        afrag[t].x = hi ? a2  : a0;
        afrag[t].y = hi ? 0.f : a1;
    }

    // ---- Running top-16 (ascending) per lane; static indexing only -> VGPRs.
    float dist[KSEL];
    int   idx[KSEL];
#pragma unroll
    for (int j = 0; j < KSEL; ++j) { dist[j] = __builtin_huge_valf(); idx[j] = 0; }

    float* const mykeys = &keybuf[w][0];

    // ---- Software pipeline: preload candidate-tile 0 coordinates.
    float cy0 = xb[nc * 3 + 0];
    float cy1 = xb[nc * 3 + 1];
    float cy2 = xb[nc * 3 + 2];

    for (int cb = 0; cb < NPTS; cb += 16) {
        // ---- B fragment (4x16 f32) for columns cb..cb+15, + |y|^2 per column.
        const float sqc = fmaf(cy0, cy0, fmaf(cy1, cy1, cy2 * cy2));
        v2f bfrag;
        bfrag.x = hi ? cy2 : cy0;   // K=0 (lo) / K=2 (hi)
        bfrag.y = hi ? 0.f : cy1;   // K=1 (lo) / K=3 (hi, zero pad)

        // ---- Prefetch next tile's coordinates (wraps to tile 0 at the end;
        // harmless redundant load, keeps the loop branch-free). The insertion
        // filter below hides the load latency.
        const int ncol = ((cb + 16) & (NPTS - 1)) + nc;
        const float py0 = xb[ncol * 3 + 0];
        const float py1 = xb[ncol * 3 + 1];
        const float py2 = xb[ncol * 3 + 2];

        // ---- Two 16x16 Gram tiles -> keys -> LDS transpose (row-major).
#pragma unroll
        for (int t = 0; t < 2; ++t) {
            v8f acc = {};
            acc = __builtin_amdgcn_wmma_f32_16x16x4_f32(
                /*neg_a=*/false, afrag[t], /*neg_b=*/false, bfrag,
                /*c_mod=*/(short)0, acc, /*reuse_a=*/false, /*reuse_b=*/false);
            // D layout: lanes 0-15 hold (M=v, N=lane); lanes 16-31 (M=v+8, N=lane-16)
#pragma unroll
            for (int v = 0; v < 8; ++v) {
                const float key = fmaf(-2.0f, acc[v], sqc);
                mykeys[(16 * t + v + 8 * hi) * 16 + nc] = key;
            }
        }

        // LDS ops from the same wave are in-order: safe to read back without barrier.
        float cand[16];
        {
            const float4* rp = (const float4*)(mykeys + lane * 16);
            const float4 q0 = rp[0], q1 = rp[1], q2 = rp[2], q3 = rp[3];
            cand[0] = q0.x; cand[1] = q0.y; cand[2]  = q0.z; cand[3]  = q0.w;
            cand[4] = q1.x; cand[5] = q1.y; cand[6]  = q1.z; cand[7]  = q1.w;
            cand[8] = q2.x; cand[9] = q2.y; cand[10] = q2.z; cand[11] = q2.w;
            cand[12] = q3.x; cand[13] = q3.y; cand[14] = q3.z; cand[15] = q3.w;
        }

        // ---- Tile-level guard: if even the smallest candidate in this tile
        // can't enter the current top-16, skip all 16 insertion checks.
        // (15 fminf ops, dual-issue / min3-friendly; in steady state almost
        // every tile takes this fast path.)
        float m0 = fminf(fminf(cand[0],  cand[1]),  fminf(cand[2],  cand[3]));
        float m1 = fminf(fminf(cand[4],  cand[5]),  fminf(cand[6],  cand[7]));
        float m2 = fminf(fminf(cand[8],  cand[9]),  fminf(cand[10], cand[11]));
        float m3 = fminf(fminf(cand[12], cand[13]), fminf(cand[14], cand[15]));
        const float tmin = fminf(fminf(m0, m1), fminf(m2, m3));

        if (tmin < dist[KSEL - 1]) {
            // ---- Branchless sorted insertion (fully unrolled); per-candidate
            // guard still rejects most lanes cheaply.
#pragma unroll
            for (int j = 0; j < 16; ++j) {
                const float dcur = cand[j];
                const int   ic   = cb + j;
                if (dcur < dist[KSEL - 1]) {
#pragma unroll
                    for (int s = KSEL - 1; s >= 1; --s) {
                        const bool  c  = dcur < dist[s];
                        const bool  cp = dcur < dist[s - 1];
                        const float nd = cp ? dist[s - 1] : dcur;
                        const int   ni = cp ? idx[s - 1]  : ic;
                        dist[s] = c ? nd : dist[s];
                        idx[s]  = c ? ni : idx[s];
                    }
                    if (dcur < dist[0]) { dist[0] = dcur; idx[0] = ic; }
                }
            }
        }

        // ---- Rotate the pipeline registers.
        cy0 = py0; cy1 = py1; cy2 = py2;
    }

    // ---- Dilation: emit indices 0,2,...,14 of the ascending list.
    const int gout = growbase + lane;                 // global row (b*NPTS + r)
    int4* op = (int4*)(out + (size_t)gout * KOUT);
    op[0] = make_int4(idx[0], idx[2],  idx[4],  idx[6]);
    op[1] = make_int4(idx[8], idx[10], idx[12], idx[14]);
}

extern "C" void kernel_launch(void* const* d_in, const int* in_sizes, int n_in,
                              void* d_out, int out_size, void* d_ws, size_t ws_size,
                              hipStream_t stream) {
    (void)in_sizes; (void)n_in; (void)d_ws; (void)ws_size; (void)out_size;
    const float* x = (const float*)d_in[0];   // (4, 8192, 3) f32
    int* out = (int*)d_out;                   // (4, 8192, 8) i32
    const int total_rows = 4 * NPTS;
    const int blocks = total_rows / ROWS_PER_BLOCK;   // 128
    dilated_knn_wmma_kernel<<<blocks, 256, 0, stream>>>(x, out);
}